// UpSampling_34265249088350
// MI455X (gfx1250) — compile-verified
//
#include <hip/hip_runtime.h>
#include <float.h>

typedef float v2f __attribute__((ext_vector_type(2)));
typedef float v4f __attribute__((ext_vector_type(4)));
typedef float v8f __attribute__((ext_vector_type(8)));

#define B_     4
#define N1_    4096
#define N2_    16384
#define CDEC   512
#define CENC   256
#define COUT   256
#define KNN_   3
#define BN_EPS 1e-5f
#define FP_EPS 1e-8f

// V_WMMA_F32_16X16X4_F32: D(16x16 f32, 8 VGPR) = A(16x4 f32, 2 VGPR) x B(4x16 f32, 2 VGPR) + C
static __device__ __forceinline__ v8f wmma4(v2f a, v2f b, v8f c) {
  return __builtin_amdgcn_wmma_f32_16x16x4_f32(
      /*neg_a=*/false, a, /*neg_b=*/false, b,
      /*c_mod=*/(short)0, c, /*reuse_a=*/false, /*reuse_b=*/false);
}

// ---------------------------------------------------------------------------
// Kernel 1: feats1 = relu(bn(points1 @ W1^T + b1))   M=B*N1, K=CDEC, N=COUT
// BN folded into per-channel scale/shift: y = relu(dot*sc[c] + sh[c])
//   sc = g * rsqrt(rv+eps), sh = (bias - rm)*sc + beta
// Wave tile: 16 rows x 64 cols (4 v8f accumulators share one A fragment).
// Block: 8 waves = 32 rows x 256 cols.
// ---------------------------------------------------------------------------
template <int K>
__global__ __launch_bounds__(256) void gemm_bn_relu_k(
    const float* __restrict__ A, const float* __restrict__ W,
    const float* __restrict__ bias, const float* __restrict__ gamma,
    const float* __restrict__ beta, const float* __restrict__ rmean,
    const float* __restrict__ rvar, float* __restrict__ out) {
  const int tid  = threadIdx.x;
  const int wave = tid >> 5;
  const int lane = tid & 31;
  const int half = lane >> 4;   // selects K pair within fragment
  const int l15  = lane & 15;   // row (A) / col (B,D) within 16
  const int m0 = (blockIdx.x * 2 + (wave >> 2)) * 16;
  const int n0 = (wave & 3) * 64;

  // A fragment: lane needs A[m0+l15][kk + 2*half + {0,1}] -> one float2 load
  const float* aRow  = A + (size_t)(m0 + l15) * K + 2 * half;
  // B fragment for col-tile j: B[k][n] = W[n][k] -> float2 from W row n0+16j+l15
  const float* bRow0 = W + (size_t)(n0 +  0 + l15) * K + 2 * half;
  const float* bRow1 = W + (size_t)(n0 + 16 + l15) * K + 2 * half;
  const float* bRow2 = W + (size_t)(n0 + 32 + l15) * K + 2 * half;
  const float* bRow3 = W + (size_t)(n0 + 48 + l15) * K + 2 * half;

  v8f z = {};
  v8f acc[4] = {z, z, z, z};
  #pragma unroll 4
  for (int kk = 0; kk < K; kk += 4) {
    v2f a  = *(const v2f*)(aRow  + kk);
    v2f b0 = *(const v2f*)(bRow0 + kk);
    v2f b1 = *(const v2f*)(bRow1 + kk);
    v2f b2 = *(const v2f*)(bRow2 + kk);
    v2f b3 = *(const v2f*)(bRow3 + kk);
    acc[0] = wmma4(a, b0, acc[0]);
    acc[1] = wmma4(a, b1, acc[1]);
    acc[2] = wmma4(a, b2, acc[2]);
    acc[3] = wmma4(a, b3, acc[3]);
  }

  float sc[4], sh[4];
  #pragma unroll
  for (int j = 0; j < 4; ++j) {
    const int c = n0 + 16 * j + l15;   // D cols depend only on lane&15
    const float s = gamma[c] * rsqrtf(rvar[c] + BN_EPS);
    sc[j] = s;
    sh[j] = (bias[c] - rmean[c]) * s + beta[c];
  }
  #pragma unroll
  for (int r = 0; r < 8; ++r) {
    const int row = m0 + r + 8 * half;
    #pragma unroll
    for (int j = 0; j < 4; ++j) {
      const int col = n0 + 16 * j + l15;
      const float v = fmaxf(acc[j][r] * sc[j] + sh[j], 0.0f);
      out[(size_t)row * COUT + col] = v;
    }
  }
}

// ---------------------------------------------------------------------------
// Kernel 2: 3-NN over xyz1 per query in xyz2, reference metric
//   d = |x2|^2 + |x1|^2 - 2*dot.
// xyz1[b] staged in LDS as float4 {x, y, z, |p|^2}: one ds_load_b128 per
// candidate (4x fewer DS issues than scalar arrays), 64 KB LDS, and all
// lanes read the same address in lockstep -> pure broadcast, no conflicts.
// One block per (batch, 256-query chunk); scan unrolled x4 so DS loads
// clause and overlap the compare/insert chain.
// ---------------------------------------------------------------------------
__global__ __launch_bounds__(256) void knn3_weights(
    const float* __restrict__ xyz1, const float* __restrict__ xyz2,
    int* __restrict__ nidx, float* __restrict__ nw) {
  __shared__ v4f sp[N1_];              // {x, y, z, |p|^2}  = 64 KB
  const int tid   = threadIdx.x;
  const int b     = blockIdx.x >> 6;   // 64 chunks per batch
  const int chunk = blockIdx.x & 63;

  for (int i = tid; i < N1_; i += 256) {
    const size_t o = ((size_t)b * N1_ + i) * 3;
    const float x = xyz1[o + 0], y = xyz1[o + 1], zc = xyz1[o + 2];
    v4f p;
    p.x = x; p.y = y; p.z = zc;
    p.w = x * x + y * y + zc * zc;
    sp[i] = p;
  }
  __syncthreads();

  const int q = b * N2_ + chunk * 256 + tid;
  const float qx = xyz2[(size_t)q * 3 + 0];
  const float qy = xyz2[(size_t)q * 3 + 1];
  const float qz = xyz2[(size_t)q * 3 + 2];
  const float q2 = qx * qx + qy * qy + qz * qz;
  const float nqx = -2.0f * qx, nqy = -2.0f * qy, nqz = -2.0f * qz;

  float d0 = FLT_MAX, d1 = FLT_MAX, d2 = FLT_MAX;
  int   i0 = 0, i1 = 0, i2 = 0;
  #pragma unroll 4
  for (int j = 0; j < N1_; ++j) {
    const v4f p = sp[j];
    // d = q2 + |p|^2 - 2*dot  (matches reference numerics)
    const float d = fmaf(nqx, p.x, fmaf(nqy, p.y, fmaf(nqz, p.z, q2 + p.w)));
    if (d < d2) {
      if (d < d1) {
        d2 = d1; i2 = i1;
        if (d < d0) { d1 = d0; i1 = i0; d0 = d; i0 = j; }
        else        { d1 = d;  i1 = j; }
      } else { d2 = d; i2 = j; }
    }
  }
  const float r0 = 1.0f / (d0 + FP_EPS);
  const float r1 = 1.0f / (d1 + FP_EPS);
  const float r2 = 1.0f / (d2 + FP_EPS);
  const float inv = 1.0f / (r0 + r1 + r2);
  nidx[(size_t)q * 3 + 0] = i0;  nw[(size_t)q * 3 + 0] = r0 * inv;
  nidx[(size_t)q * 3 + 1] = i1;  nw[(size_t)q * 3 + 1] = r1 * inv;
  nidx[(size_t)q * 3 + 2] = i2;  nw[(size_t)q * 3 + 2] = r2 * inv;
}

// ---------------------------------------------------------------------------
// Kernel 3: out2 = relu(bn(points2 @ W2^T + b2)) + interp(feats1, idx, w)
// Same WMMA structure as kernel 1 (K=CENC); interp fused into the epilogue:
// 3 gathers/element, coalesced across the 16 lanes sharing a D row.
// ---------------------------------------------------------------------------
template <int K>
__global__ __launch_bounds__(256) void gemm_bn_relu_interp_k(
    const float* __restrict__ A, const float* __restrict__ W,
    const float* __restrict__ bias, const float* __restrict__ gamma,
    const float* __restrict__ beta, const float* __restrict__ rmean,
    const float* __restrict__ rvar,
    const float* __restrict__ feats1, const int* __restrict__ nidx,
    const float* __restrict__ nw, float* __restrict__ out2) {
  const int tid  = threadIdx.x;
  const int wave = tid >> 5;
  const int lane = tid & 31;
  const int half = lane >> 4;
  const int l15  = lane & 15;
  const int m0 = (blockIdx.x * 2 + (wave >> 2)) * 16;
  const int n0 = (wave & 3) * 64;

  const float* aRow  = A + (size_t)(m0 + l15) * K + 2 * half;
  const float* bRow0 = W + (size_t)(n0 +  0 + l15) * K + 2 * half;
  const float* bRow1 = W + (size_t)(n0 + 16 + l15) * K + 2 * half;
  const float* bRow2 = W + (size_t)(n0 + 32 + l15) * K + 2 * half;
  const float* bRow3 = W + (size_t)(n0 + 48 + l15) * K + 2 * half;

  v8f z = {};
  v8f acc[4] = {z, z, z, z};
  #pragma unroll 4
  for (int kk = 0; kk < K; kk += 4) {
    v2f a  = *(const v2f*)(aRow  + kk);
    v2f b0 = *(const v2f*)(bRow0 + kk);
    v2f b1 = *(const v2f*)(bRow1 + kk);
    v2f b2 = *(const v2f*)(bRow2 + kk);
    v2f b3 = *(const v2f*)(bRow3 + kk);
    acc[0] = wmma4(a, b0, acc[0]);
    acc[1] = wmma4(a, b1, acc[1]);
    acc[2] = wmma4(a, b2, acc[2]);
    acc[3] = wmma4(a, b3, acc[3]);
  }

  float sc[4], sh[4];
  #pragma unroll
  for (int j = 0; j < 4; ++j) {
    const int c = n0 + 16 * j + l15;
    const float s = gamma[c] * rsqrtf(rvar[c] + BN_EPS);
    sc[j] = s;
    sh[j] = (bias[c] - rmean[c]) * s + beta[c];
  }

  #pragma unroll
  for (int r = 0; r < 8; ++r) {
    const int row = m0 + r + 8 * half;          // flat index over B*N2
    const int b   = row >> 14;                  // row / N2_
    const int ia = nidx[(size_t)row * 3 + 0];
    const int ib = nidx[(size_t)row * 3 + 1];
    const int ic = nidx[(size_t)row * 3 + 2];
    const float w0 = nw[(size_t)row * 3 + 0];
    const float w1 = nw[(size_t)row * 3 + 1];
    const float w2 = nw[(size_t)row * 3 + 2];
    const float* f1b = feats1 + (size_t)b * N1_ * COUT;
    #pragma unroll
    for (int j = 0; j < 4; ++j) {
      const int col = n0 + 16 * j + l15;
      float v = fmaxf(acc[j][r] * sc[j] + sh[j], 0.0f);
      v = fmaf(w0, f1b[(size_t)ia * COUT + col],
          fmaf(w1, f1b[(size_t)ib * COUT + col],
          fmaf(w2, f1b[(size_t)ic * COUT + col], v)));
      out2[(size_t)row * COUT + col] = v;
    }
  }
}

extern "C" void kernel_launch(void* const* d_in, const int* in_sizes, int n_in,
                              void* d_out, int out_size, void* d_ws, size_t ws_size,
                              hipStream_t stream) {
  (void)in_sizes; (void)n_in; (void)out_size; (void)ws_size;
  const float* xyz1    = (const float*)d_in[0];
  const float* points1 = (const float*)d_in[1];
  const float* xyz2    = (const float*)d_in[2];
  const float* points2 = (const float*)d_in[3];
  const float* W1  = (const float*)d_in[4];
  const float* b1  = (const float*)d_in[5];
  const float* g1  = (const float*)d_in[6];
  const float* be1 = (const float*)d_in[7];
  const float* rm1 = (const float*)d_in[8];
  const float* rv1 = (const float*)d_in[9];
  const float* W2  = (const float*)d_in[10];
  const float* b2  = (const float*)d_in[11];
  const float* g2  = (const float*)d_in[12];
  const float* be2 = (const float*)d_in[13];
  const float* rm2 = (const float*)d_in[14];
  const float* rv2 = (const float*)d_in[15];

  // Workspace: feats1 (B*N1*COUT floats = 16 MiB), then knn idx + weights
  float* feats1 = (float*)d_ws;
  const size_t feats1_elems = (size_t)B_ * N1_ * COUT;        // 4,194,304
  int*   nidx = (int*)((char*)d_ws + feats1_elems * sizeof(float));
  float* nw   = (float*)((char*)nidx + (size_t)B_ * N2_ * KNN_ * sizeof(int));

  // Output tuple: xyz2 flat (B*N2*3) || interp+feats2 (B*N2*COUT)
  float* outp = (float*)d_out;
  float* out2 = outp + (size_t)B_ * N2_ * 3;

  hipMemcpyAsync(outp, xyz2, (size_t)B_ * N2_ * 3 * sizeof(float),
                 hipMemcpyDeviceToDevice, stream);

  gemm_bn_relu_k<CDEC><<<(B_ * N1_) / 32, 256, 0, stream>>>(
      points1, W1, b1, g1, be1, rm1, rv1, feats1);

  knn3_weights<<<B_ * (N2_ / 256), 256, 0, stream>>>(xyz1, xyz2, nidx, nw);

  gemm_bn_relu_interp_k<CENC><<<(B_ * N2_) / 32, 256, 0, stream>>>(
      points2, W2, b2, g2, be2, rm2, rv2, feats1, nidx, nw, out2);
}